// NerTr_34608846471269
// MI455X (gfx1250) — compile-verified
//
#include <hip/hip_runtime.h>
#include <math.h>

#define Bk 32
#define Tk 1024
#define Dk 768
#define Wk 400
#define Nk 9
#define APAD 4                 // pad A rows to 772 floats (16B-aligned rows, fewer bank conflicts)
#define TSPLIT 8               // T-dimension split for the segment-max scan
#define TCHUNK (Tk / TSPLIT)
#define DCH (Dk / 256)         // d-chunks per batch in seg-max

typedef __attribute__((ext_vector_type(2))) float v2f;
typedef __attribute__((ext_vector_type(8))) float v8f;
typedef __attribute__((ext_vector_type(4))) int   v4i;

// exact parameter types for the async-LDS builtin (AS1/AS3 int4*), via inttoptr
#define GAS1 __attribute__((address_space(1)))
#define LAS3 __attribute__((address_space(3)))
#define G_V4IP(p) ((GAS1 v4i*)(uintptr_t)(p))
#define L_V4IP(p) ((LAS3 v4i*)(uint32_t)(uintptr_t)(p))

// ---------------------------------------------------------------- fill pooled with -inf (max identity)
__global__ void fill_neginf_kernel(float4* __restrict__ p, int n4) {
    const int i = blockIdx.x * blockDim.x + threadIdx.x;
    const float ninf = __int_as_float(0xff800000u);
    if (i < n4) p[i] = make_float4(ninf, ninf, ninf, ninf);
}

// ---------------------------------------------------------------- normalize the 9 query rows
__global__ void norm_queries_kernel(const float* __restrict__ q, float* __restrict__ qn) {
    __shared__ float red[256];
    const int n = blockIdx.x;
    const int tid = threadIdx.x;
    float ss = 0.f;
    for (int d = tid; d < Dk; d += 256) {
        const float x = q[n * Dk + d];
        ss += x * x;
    }
    red[tid] = ss;
    __syncthreads();
    for (int s = 128; s > 0; s >>= 1) {
        if (tid < s) red[tid] += red[tid + s];
        __syncthreads();
    }
    const float scale = 1.0f / (sqrtf(red[0]) + 1e-8f);
    for (int d = tid; d < Dk; d += 256)
        qn[n * Dk + d] = q[n * Dk + d] * scale;
}

// ---------------------------------------------------------------- segment max over sorted ids
// Grid: B * DCH * TSPLIT blocks of 256. Each block scans a T-chunk keeping a running max
// and flushes on word-id change via non-returning float atomic-max (seam-safe).
__global__ void seg_max_kernel(const float* __restrict__ h,
                               const int*   __restrict__ ids,
                               float*       __restrict__ pooled) {
    const int blk = blockIdx.x;
    const int ts = blk % TSPLIT;
    const int c  = (blk / TSPLIT) % DCH;
    const int b  = blk / (TSPLIT * DCH);
    const int d  = c * 256 + threadIdx.x;
    const float* hb  = h + (size_t)b * Tk * Dk + d;
    const int*   idb = ids + b * Tk;
    const int t0 = ts * TCHUNK, t1 = t0 + TCHUNK;
    float m = 0.f;
    int prev = -1;
    for (int t = t0; t < t1; ++t) {
        const int   w = idb[t];                      // uniform -> scalar load
        const float v = hb[(size_t)t * Dk];
        const int tp = (t + 8 < t1) ? (t + 8) : (t1 - 1);
        __builtin_prefetch(hb + (size_t)tp * Dk, 0, 1);
        if (w != prev) {
            if (prev >= 0) atomicMax(&pooled[((size_t)b * Wk + prev) * Dk + d], m);
            prev = w;
            m = v;
        } else {
            m = fmaxf(m, v);
        }
    }
    if (prev >= 0) atomicMax(&pooled[((size_t)b * Wk + prev) * Dk + d], m);
}

// ---------------------------------------------------------------- LayerNorm + cosine (WMMA) + softmax
__global__ __launch_bounds__(256) void ner_tile_kernel(
        const float* __restrict__ pooled, const float* __restrict__ gamma,
        const float* __restrict__ beta,   const float* __restrict__ qn,
        float* __restrict__ out) {
    __shared__ float A[16][Dk + APAD];     // 16 words x 768 dims (normed in place)
    __shared__ float part[8 * 256];        // per-wave WMMA partial accumulators
    __shared__ float rnorm_s[16];
    __shared__ float cosm[16][16];

    const int tid = threadIdx.x;
    const int b  = blockIdx.x / (Wk / 16);
    const int w0 = (blockIdx.x % (Wk / 16)) * 16;

    // ---- tile load: async global->LDS (ASYNCcnt) when available, float4 path otherwise
    const float4* src = (const float4*)(pooled + ((size_t)b * Wk + w0) * Dk);
#if __has_builtin(__builtin_amdgcn_global_load_async_to_lds_b128)
    for (int i = tid; i < 16 * Dk / 4; i += 256) {
        const int row = i / (Dk / 4);
        const int c4  = i % (Dk / 4);
        __builtin_amdgcn_global_load_async_to_lds_b128(
            G_V4IP(src + i), L_V4IP(&A[row][0] + 4 * c4), 0, 0);
    }
#else
    for (int i = tid; i < 16 * Dk / 4; i += 256) {
        const int row = i / (Dk / 4);
        const int c4  = i % (Dk / 4);
        *(((float4*)&A[row][0]) + c4) = src[i];
    }
#endif

    // ---- preload the loop-invariant B fragments (query rows) while the copy is in flight
    const int wave = tid >> 5;
    const int lane = tid & 31;
    const int hi   = lane >> 4;          // lanes 16-31 carry K+2,K+3
    const int l16  = lane & 15;          // row (A) / col n (B)
    const int   qr     = (l16 < Nk) ? l16 : 0;
    const float bscale = (l16 < Nk) ? 1.f : 0.f;   // zero cols 9..15 once, EXEC stays all-ones
    const float* qrow  = qn + qr * Dk;
    const int kbase = wave * (Dk / 8) + 2 * hi;    // 96-wide K chunk per wave
    v2f bfr[24];
#pragma unroll
    for (int i = 0; i < 24; ++i) {
        const int k = kbase + 4 * i;
        bfr[i].x = qrow[k]     * bscale;
        bfr[i].y = qrow[k + 1] * bscale;
    }

#if __has_builtin(__builtin_amdgcn_s_wait_asynccnt)
    __builtin_amdgcn_s_wait_asynccnt(0);
#else
    asm volatile("s_wait_asynccnt 0x0" ::: "memory");
#endif
    __syncthreads();

    // ---- per-row LayerNorm stats: 16 lanes per row, xor-butterfly (wave32, width 16)
    // -inf sentinel (empty word) -> 0, matching the reference's empty-segment value.
    const int r = tid >> 4;
    const int j = tid & 15;
    float s = 0.f, ss = 0.f;
    for (int d = j; d < Dk; d += 16) {
        float x = A[r][d];
        x = (x < -1e37f) ? 0.f : x;
        s  += x;
        ss += x * x;
    }
    for (int off = 8; off > 0; off >>= 1) {
        s  += __shfl_xor(s,  off, 16);
        ss += __shfl_xor(ss, off, 16);
    }
    const float mu   = s * (1.0f / Dk);
    const float var  = ss * (1.0f / Dk) - mu * mu;
    const float istd = rsqrtf(var + 1e-5f);

    // ---- normalize in place + accumulate ||normed||^2
    float ns = 0.f;
    for (int d = j; d < Dk; d += 16) {
        float x0 = A[r][d];
        x0 = (x0 < -1e37f) ? 0.f : x0;
        const float x = (x0 - mu) * istd * gamma[d] + beta[d];
        A[r][d] = x;
        ns += x * x;
    }
    for (int off = 8; off > 0; off >>= 1) ns += __shfl_xor(ns, off, 16);
    if (j == 0) rnorm_s[r] = sqrtf(ns) + 1e-8f;
    __syncthreads();

    // ---- WMMA f32 16x16x4: D[16w x 16n] = A[16x768] * qn^T ; K split across 8 waves
    v8f acc = {};
#pragma unroll
    for (int i = 0; i < 24; ++i) {
        const int k = kbase + 4 * i;
        v2f a;
        a.x = A[l16][k];
        a.y = A[l16][k + 1];
        acc = __builtin_amdgcn_wmma_f32_16x16x4_f32(
                  false, a, false, bfr[i], (short)0, acc, false, false);
    }

    // ---- reduce the 8 per-wave partial C tiles through LDS
    {
        float* pw = &part[wave * 256 + lane * 8];
#pragma unroll
        for (int v = 0; v < 8; ++v) pw[v] = acc[v];
    }
    __syncthreads();
    float sum = 0.f;
#pragma unroll
    for (int w8 = 0; w8 < 8; ++w8) sum += part[w8 * 256 + tid];
    // slot tid = lane*8 + v ; C layout: m = v + 8*(lane>=16), n = lane&15
    const int plane = tid >> 3;
    const int pv    = tid & 7;
    const int m = pv + 8 * (plane >> 4);
    const int n = plane & 15;
    cosm[m][n] = sum / rnorm_s[m];
    __syncthreads();

    // ---- softmax over the 9 classes, one word per thread
    if (tid < 16) {
        float mx = cosm[tid][0];
#pragma unroll
        for (int nn = 1; nn < Nk; ++nn) mx = fmaxf(mx, cosm[tid][nn]);
        float e[Nk];
        float se = 0.f;
#pragma unroll
        for (int nn = 0; nn < Nk; ++nn) { e[nn] = __expf(cosm[tid][nn] - mx); se += e[nn]; }
        const float inv = 1.0f / se;
        float* o = out + ((size_t)b * Wk + w0 + tid) * Nk;
#pragma unroll
        for (int nn = 0; nn < Nk; ++nn) o[nn] = e[nn] * inv;
    }
}

// ----------------------------------------------------------------
extern "C" void kernel_launch(void* const* d_in, const int* in_sizes, int n_in,
                              void* d_out, int out_size, void* d_ws, size_t ws_size,
                              hipStream_t stream) {
    const float* h     = (const float*)d_in[0];   // [B,T,D]
    const float* gamma = (const float*)d_in[1];   // [D]
    const float* beta  = (const float*)d_in[2];   // [D]
    const float* q     = (const float*)d_in[3];   // [N,D]
    const int*   ids   = (const int*)d_in[4];     // [B,T]
    float* out = (float*)d_out;                   // [B,W,N]

    float* pooled = (float*)d_ws;                      // B*W*D floats (39.3 MB)
    float* qn     = pooled + (size_t)Bk * Wk * Dk;     // N*D floats

    const int n4 = Bk * Wk * Dk / 4;
    fill_neginf_kernel<<<(n4 + 255) / 256, 256, 0, stream>>>((float4*)pooled, n4);
    norm_queries_kernel<<<Nk, 256, 0, stream>>>(q, qn);
    seg_max_kernel<<<Bk * DCH * TSPLIT, 256, 0, stream>>>(h, ids, pooled);
    ner_tile_kernel<<<Bk * (Wk / 16), 256, 0, stream>>>(pooled, gamma, beta, qn, out);
}